// GCN_76836964926204
// MI455X (gfx1250) — compile-verified
//
#include <hip/hip_runtime.h>
#include <math.h>

typedef float v2f  __attribute__((ext_vector_type(2)));
typedef float v8f  __attribute__((ext_vector_type(8)));

constexpr int NN   = 131072;
constexpr int EE   = 1048576;
constexpr int HH   = 128;
constexpr int OUTD = 40;
constexpr float BN_EPS = 1e-5f;
constexpr float RESW   = 0.2f;

// ---------------------------------------------------------------------------
// GEMM: Y[N x 128] = X[N x 128] @ W[128 x 128] (+ bias per column, optional)
// Block: 128 threads = 4 waves; each wave owns a 16-row strip x all 128 cols
// (8 accumulator tiles of 16x16), fp32 WMMA 16x16x4, K-loop of 32 steps.
// W staged in LDS in K-paired layout: sW2[(k/2)*256 + col*2 + (k&1)] so each
// lane's (K,K+1) B-fragment is one ds_load_b64 (no register repacking).
// ---------------------------------------------------------------------------
__global__ __launch_bounds__(128) void gcn_gemm128(
    const float* __restrict__ X, const float* __restrict__ W,
    const float* __restrict__ bias, float* __restrict__ Y)
{
  __shared__ float sW2[128 * 128];
  const int tid = threadIdx.x;
  // cooperative stage: each thread handles 32 float4 rows-chunks of W,
  // scattering into the K-paired layout.
  for (int i = tid * 4; i < 128 * 128; i += 128 * 4) {
    const int k = i >> 7;          // source row
    const int c = i & 127;         // source col (multiple of 4)
    const float4 v = *(const float4*)&W[i];
    float* d = &sW2[(k >> 1) * 256 + c * 2 + (k & 1)];
    d[0] = v.x; d[2] = v.y; d[4] = v.z; d[6] = v.w;
  }
  __syncthreads();

  const int wave = tid >> 5;
  const int lane = tid & 31;
  const int row0 = (blockIdx.x * 4 + wave) * 16;
  const int mrow = lane & 15;
  const int hi   = lane >> 4;            // 0: lanes 0-15, 1: lanes 16-31
  const int khalf = hi * 2;              // lanes 16-31 carry K+2, K+3

  v8f acc[8];
  const v8f vzero = {0.f, 0.f, 0.f, 0.f, 0.f, 0.f, 0.f, 0.f};
#pragma unroll
  for (int j = 0; j < 8; ++j) acc[j] = vzero;

  const float* xrow = X + (size_t)(row0 + mrow) * 128;

  for (int k = 0; k < 128; k += 4) {
    // A fragment: one 8-byte load (k + khalf is even, row base 512B aligned)
    const v2f a = *(const v2f*)&xrow[k + khalf];
    const float* bbase = &sW2[((k + khalf) >> 1) * 256];
#pragma unroll
    for (int j = 0; j < 8; ++j) {
      const int col = j * 16 + mrow;
      const v2f b = *(const v2f*)&bbase[col * 2];   // ds_load_b64
      acc[j] = __builtin_amdgcn_wmma_f32_16x16x4_f32(
          false, a, false, b, (short)0, acc[j], false, false);
    }
  }

  // store: VGPR r of a tile holds rows (row0 + hi*8 + r), column j*16 + mrow
  const int rbase = row0 + hi * 8;
#pragma unroll
  for (int j = 0; j < 8; ++j) {
    const int col = j * 16 + mrow;
    const float bv = bias ? bias[col] : 0.f;
#pragma unroll
    for (int r = 0; r < 8; ++r)
      Y[(size_t)(rbase + r) * 128 + col] = acc[j][r] + bv;
  }
}

// ---------------------------------------------------------------------------
// agg[n][c] = conv_b[c]; block 0 additionally zeroes the BN stat accumulators
// ---------------------------------------------------------------------------
__global__ __launch_bounds__(256) void gcn_init_agg(
    float* __restrict__ agg, const float* __restrict__ cb, float* __restrict__ stats)
{
  const size_t i = (size_t)blockIdx.x * 256 + threadIdx.x;
  agg[i] = cb[i & 127];
  if (blockIdx.x == 0) stats[threadIdx.x] = 0.f;  // 256 floats: sum + sumsq
}

// ---------------------------------------------------------------------------
// one wave per edge: gather h[src] (float4/lane), scale, atomic-add to agg[dst]
// ---------------------------------------------------------------------------
__global__ __launch_bounds__(256) void gcn_scatter(
    const float* __restrict__ h, const int* __restrict__ ei,
    const float* __restrict__ ew, float* __restrict__ agg)
{
  const int e    = blockIdx.x * 8 + (threadIdx.x >> 5);
  const int lane = threadIdx.x & 31;
  const int src = ei[e];
  const int dst = ei[EE + e];
  const float w = ew[e];
  const float4 v = *(const float4*)&h[(size_t)src * 128 + lane * 4];
  float* p = &agg[(size_t)dst * 128 + lane * 4];
  atomicAdd(p + 0, v.x * w);
  atomicAdd(p + 1, v.y * w);
  atomicAdd(p + 2, v.z * w);
  atomicAdd(p + 3, v.w * w);
}

// ---------------------------------------------------------------------------
// per-channel sum / sum-of-squares over the node dim (LDS partials + atomics)
// ---------------------------------------------------------------------------
__global__ __launch_bounds__(256) void gcn_bn_stats(
    const float* __restrict__ agg, float* __restrict__ stats, int rowsPerBlock)
{
  const int c  = threadIdx.x & 127;
  const int rh = threadIdx.x >> 7;       // 0 or 1
  const int r0 = blockIdx.x * rowsPerBlock;
  float s = 0.f, q = 0.f;
  for (int r = r0 + rh; r < r0 + rowsPerBlock; r += 2) {
    const float v = agg[(size_t)r * 128 + c];
    s += v;
    q += v * v;
  }
  __shared__ float ls[256], lq[256];
  ls[threadIdx.x] = s;
  lq[threadIdx.x] = q;
  __syncthreads();
  if (threadIdx.x < 128) {
    atomicAdd(&stats[c],       ls[threadIdx.x] + ls[threadIdx.x + 128]);
    atomicAdd(&stats[128 + c], lq[threadIdx.x] + lq[threadIdx.x + 128]);
  }
}

// ---------------------------------------------------------------------------
// fused: BN (batch stats) -> ReLU -> +0.2*inp -> x (in place over agg),
// and acc (+)= softmax(layer_weights)[layer] * x
// ---------------------------------------------------------------------------
__global__ __launch_bounds__(256) void gcn_bn_relu_res(
    float* __restrict__ x, const float* __restrict__ inp, float* __restrict__ acc,
    const float* __restrict__ stats, const float* __restrict__ gamma,
    const float* __restrict__ beta, const float* __restrict__ lw, int layer)
{
  const size_t i = (size_t)blockIdx.x * 256 + threadIdx.x;
  const int c = (int)(i & 127);
  const float invN = 1.f / (float)NN;
  const float mu  = stats[c] * invN;
  const float var = stats[128 + c] * invN - mu * mu;
  const float sc  = gamma[c] * rsqrtf(var + BN_EPS);
  float v = (x[i] - mu) * sc + beta[c];
  v = fmaxf(v, 0.f);
  v += RESW * inp[i];
  x[i] = v;

  const float l0 = lw[0], l1 = lw[1], l2 = lw[2];
  const float m  = fmaxf(l0, fmaxf(l1, l2));
  const float e0 = __expf(l0 - m), e1 = __expf(l1 - m), e2 = __expf(l2 - m);
  const float sel = (layer == 0) ? e0 : (layer == 1) ? e1 : e2;
  const float s = sel / (e0 + e1 + e2);
  if (layer == 0) acc[i] = s * v;
  else            acc[i] += s * v;
}

// ---------------------------------------------------------------------------
// fused final: out = log_softmax(X @ lin_W + lin_b); one wave per row
// ---------------------------------------------------------------------------
__global__ __launch_bounds__(256) void gcn_lin_logsoftmax(
    const float* __restrict__ X, const float* __restrict__ W,
    const float* __restrict__ b, float* __restrict__ out)
{
  __shared__ float sx[8 * 128];
  __shared__ float sv[8 * 48];
  const int wave = threadIdx.x >> 5;
  const int lane = threadIdx.x & 31;
  const int n = blockIdx.x * 8 + wave;

  *(float4*)&sx[wave * 128 + lane * 4] = *(const float4*)&X[(size_t)n * 128 + lane * 4];
  __syncthreads();

  const int o2 = 32 + (lane & 7);        // valid value only for lane < 8
  float d1 = b[lane];
  float d2 = b[o2];
  for (int k = 0; k < 128; ++k) {
    const float xk = sx[wave * 128 + k];
    d1 += xk * W[k * OUTD + lane];
    d2 += xk * W[k * OUTD + o2];
  }
  sv[wave * 48 + lane] = d1;
  if (lane < 8) sv[wave * 48 + o2] = d2;
  __syncthreads();

  float mx = -1e30f;
  for (int o = 0; o < OUTD; ++o) mx = fmaxf(mx, sv[wave * 48 + o]);
  float sum = 0.f;
  for (int o = 0; o < OUTD; ++o) sum += __expf(sv[wave * 48 + o] - mx);
  const float lse = mx + __logf(sum);

  out[(size_t)n * OUTD + lane] = d1 - lse;
  if (lane < 8) out[(size_t)n * OUTD + o2] = d2 - lse;
}

// ---------------------------------------------------------------------------
extern "C" void kernel_launch(void* const* d_in, const int* in_sizes, int n_in,
                              void* d_out, int out_size, void* d_ws, size_t ws_size,
                              hipStream_t stream) {
  (void)in_sizes; (void)n_in; (void)out_size; (void)ws_size;

  const float* x            = (const float*)d_in[0];
  const int*   edge_index   = (const int*)d_in[1];
  const float* edge_weight  = (const float*)d_in[2];
  const float* inproj_W     = (const float*)d_in[3];
  const float* inproj_b     = (const float*)d_in[4];
  const float* conv_W       = (const float*)d_in[5];
  const float* conv_b       = (const float*)d_in[6];
  const float* bn_gamma     = (const float*)d_in[7];
  const float* bn_beta      = (const float*)d_in[8];
  const float* lin_W        = (const float*)d_in[9];
  const float* lin_b        = (const float*)d_in[10];
  const float* layer_w      = (const float*)d_in[11];
  float* out = (float*)d_out;

  const size_t NH = (size_t)NN * HH;     // 16.7M floats
  float* ws    = (float*)d_ws;
  float* xin   = ws;                     // projected input (residual source)
  float* A     = ws + NH;                // xcur / agg (ping)
  float* B     = ws + 2 * NH;            // h (pong)
  float* acc   = ws + 3 * NH;            // layer-weighted accumulation
  float* stats = ws + 4 * NH;            // 256 floats: per-channel sum, sumsq

  const dim3 gemmGrid(NN / 64), gemmBlk(128);
  const dim3 ewGrid((unsigned)(NH / 256)), blk256(256);

  // input projection (with bias)
  gcn_gemm128<<<gemmGrid, gemmBlk, 0, stream>>>(x, inproj_W, inproj_b, xin);

  for (int l = 0; l < 3; ++l) {
    const float* gin = (l == 0) ? xin : A;
    gcn_gemm128<<<gemmGrid, gemmBlk, 0, stream>>>(gin, conv_W + (size_t)l * HH * HH,
                                                  nullptr, B);
    gcn_init_agg<<<ewGrid, blk256, 0, stream>>>(A, conv_b + l * HH, stats);
    gcn_scatter<<<dim3(EE / 8), blk256, 0, stream>>>(B, edge_index, edge_weight, A);
    gcn_bn_stats<<<dim3(512), blk256, 0, stream>>>(A, stats, NN / 512);
    gcn_bn_relu_res<<<ewGrid, blk256, 0, stream>>>(A, xin, acc, stats,
                                                   bn_gamma + l * HH, bn_beta + l * HH,
                                                   layer_w, l);
  }

  gcn_lin_logsoftmax<<<dim3(NN / 8), blk256, 0, stream>>>(acc, lin_W, lin_b, out);
}